// LoRALayer_44143673868401
// MI455X (gfx1250) — compile-verified
//
#include <hip/hip_runtime.h>

// ---------------------------------------------------------------------------
// LoRA (per-batch rank-16) fused kernel for MI455X / gfx1250.
//   out[b] = relu(0.25 * (x[b] @ Bc_b) @ Ac_b)
//   Bc_b[d, 4j+r] = adapter_b[4b+j, d, r]   (D x 16)
//   Ac_b[4j+r, d] = adapter_a[4b+j, r, d]   (16 x D)
// Uses V_WMMA_F32_16X16X4_F32 (wave32) for both matmuls, fp32 throughout.
// ---------------------------------------------------------------------------

typedef float v2f __attribute__((ext_vector_type(2)));
typedef float v8f __attribute__((ext_vector_type(8)));

#define K_B   8
#define K_S   4096
#define K_D   2048
#define K_R   4

// D = A*B + C, fp32, 16x16x4.  (neg_a, A, neg_b, B, c_mod, C, reuse_a, reuse_b)
#define WMMA4(a, b, c) \
  __builtin_amdgcn_wmma_f32_16x16x4_f32(false, (a), false, (b), (short)0, (c), false, false)

__global__ __launch_bounds__(256)
void lora_fused_kernel(const float* __restrict__ x,    // [8][4096][2048]
                       const float* __restrict__ ab,   // [32][2048][4]
                       const float* __restrict__ aa,   // [32][4][2048]
                       float* __restrict__ out)        // [8][4096][2048]
{
  // Per-wave x staging tile: 16 rows x 32 cols, row stride 36 floats.
  // Stride 36 (row*36 mod 64 banks gives 16 distinct multiples of 4) makes the WMMA-A-layout b64 reads
  // (lane L -> row L%16, col 2*(L/16)+{0,1}) hit all 64 banks conflict-free.
  __shared__ float lds_x[8][16][36];
  // Per-wave h bounce buffer: C/D layout -> A layout conversion (16x16).
  __shared__ float lds_h[8][16][17];

  const int tid  = threadIdx.x;
  const int w    = tid >> 5;        // wave in workgroup (0..7)
  const int lane = tid & 31;

  const int wg   = blockIdx.x;      // 0..255
  const int b    = wg >> 5;         // batch (0..7)
  const int rb   = wg & 31;         // row block within batch
  const int row0 = rb * 128 + w * 16;  // this wave's first row of S

  const int mlo = lane & 15;        // N / M-low index for WMMA layouts
  const int hi  = lane >> 4;        // half-wave select
  const int khi = hi * 2;           // K offset contributed by half-wave

  const float* xw = x + ((size_t)b * K_S + row0) * K_D;

  // Step-1 B operand: Bc[d, col] with col = mlo fixed per lane.
  // Bc[d, col] = ab[(4b + col/4)*D*R + d*R + (col%4)]
  const float* bcol = ab + (size_t)(4 * b + (mlo >> 2)) * K_D * K_R + (mlo & 3);

  // -------------------------------------------------------------------------
  // Step 1: h(16x16) = x_tile(16x2048) @ Bc(2048x16), K-steps of 4 via WMMA.
  // -------------------------------------------------------------------------
  v8f acc = {0.f, 0.f, 0.f, 0.f, 0.f, 0.f, 0.f, 0.f};

  const int srow = lane >> 3;           // 0..3
  const int scol = (lane & 7) * 4;      // 0,4,...,28

  for (int kc = 0; kc < K_D; kc += 32) {
    // Stage x[row0..row0+15][kc..kc+31] -> LDS.
    // Each instr: lanes 0-7 cover one full 128B row segment -> 4 dense rows.
#pragma unroll
    for (int i = 0; i < 4; ++i) {
      const int r = i * 4 + srow;
      const float4 v = *(const float4*)(xw + (size_t)r * K_D + kc + scol);
      *(float4*)&lds_x[w][r][scol] = v;
    }
    // 8 WMMAs of K=4 over this 32-wide chunk.
#pragma unroll
    for (int c = 0; c < 8; ++c) {
      // A operand (x), b64 from LDS, conflict-free.
      v2f a = *(const v2f*)&lds_x[w][mlo][c * 4 + khi];
      // B operand (Bc), L2-resident adapter data.
      const int d = kc + c * 4 + khi;
      v2f bb;
      bb.x = bcol[(size_t)d * K_R];
      bb.y = bcol[(size_t)(d + 1) * K_R];
      acc = WMMA4(a, bb, acc);
    }
  }

  // -------------------------------------------------------------------------
  // Convert h from C/D layout to A layout via per-wave LDS bounce,
  // folding the 0.25 count/N scale.
  // -------------------------------------------------------------------------
#pragma unroll
  for (int v = 0; v < 8; ++v)
    lds_h[w][v + hi * 8][mlo] = acc[v];

  v2f hA[4];
#pragma unroll
  for (int c = 0; c < 4; ++c) {
    hA[c].x = lds_h[w][mlo][c * 4 + khi]     * 0.25f;
    hA[c].y = lds_h[w][mlo][c * 4 + khi + 1] * 0.25f;
  }

  // -------------------------------------------------------------------------
  // Step 2: out_tile(16x2048) = relu(h(16x16) @ Ac(16x2048)), 128 col blocks.
  // -------------------------------------------------------------------------
  const float* aab = aa + (size_t)(16 * b) * K_D;       // Ac base [16][2048]
  float* ow = out + ((size_t)b * K_S + row0) * K_D;

  for (int nb = 0; nb < 128; ++nb) {
    const int c0 = nb * 16 + mlo;
    v8f o = {0.f, 0.f, 0.f, 0.f, 0.f, 0.f, 0.f, 0.f};
#pragma unroll
    for (int c2 = 0; c2 < 4; ++c2) {
      const int k = c2 * 4 + khi;
      v2f bb;                                // lanes 0-15: contiguous 64B
      bb.x = aab[(size_t)k * K_D + c0];
      bb.y = aab[(size_t)(k + 1) * K_D + c0];
      o = WMMA4(hA[c2], bb, o);
    }
#pragma unroll
    for (int v = 0; v < 8; ++v) {
      const float r = o[v] > 0.f ? o[v] : 0.f;          // relu
      ow[(size_t)(v + hi * 8) * K_D + c0] = r;
    }
  }
}

extern "C" void kernel_launch(void* const* d_in, const int* in_sizes, int n_in,
                              void* d_out, int out_size, void* d_ws, size_t ws_size,
                              hipStream_t stream) {
  (void)in_sizes; (void)n_in; (void)d_ws; (void)ws_size; (void)out_size;
  const float* x  = (const float*)d_in[0];   // [8][4096][2048] f32
  const float* ab = (const float*)d_in[1];   // [32][2048][4]   f32
  const float* aa = (const float*)d_in[2];   // [32][4][2048]   f32
  float* out = (float*)d_out;                // [8][4096][2048] f32

  // 8 batches * 32 row-blocks of 128 rows; 256 threads = 8 waves per block.
  lora_fused_kernel<<<dim3(256), dim3(256), 0, stream>>>(x, ab, aa, out);
}